// AdditiveNoiseChannel_30889404792875
// MI455X (gfx1250) — compile-verified
//
#include <hip/hip_runtime.h>
#include <math.h>

// Problem constants from the reference
constexpr int Bz = 64;     // batch
constexpr int Sz = 8192;   // block size (also K of the noise GEMM)
constexpr int Tz = 129;    // taps
constexpr int Hh = 64;     // (T-1)/2

typedef float v2f __attribute__((ext_vector_type(2)));
typedef float v8f __attribute__((ext_vector_type(8)));

// ---------------------------------------------------------------------------
// Kernel 1: complex FIR ('same', zero padded), writes yr/yi into out[B,S,2].
// One block = 256 consecutive samples of one batch row; x window staged in LDS.
// ---------------------------------------------------------------------------
__global__ __launch_bounds__(256) void fir_kernel(
    const float* __restrict__ xr, const float* __restrict__ xi,
    const float* __restrict__ ar, const float* __restrict__ ai,
    float* __restrict__ out)
{
    __shared__ float sxr[384];
    __shared__ float sxi[384];
    __shared__ float sar[132];
    __shared__ float sai[132];

    const int i  = threadIdx.x;
    const int b  = blockIdx.y;
    const int n0 = blockIdx.x * 256;

    const float* xrb = xr + (size_t)b * Sz;
    const float* xib = xi + (size_t)b * Sz;

    // window covers global [n0-64, n0+319]
    int idx = n0 - Hh + i;
    bool ok = (idx >= 0) && (idx < Sz);
    sxr[i] = ok ? xrb[idx] : 0.0f;
    sxi[i] = ok ? xib[idx] : 0.0f;
    if (i < 128) {
        int idx2 = n0 + 192 + i;
        bool ok2 = idx2 < Sz;
        sxr[256 + i] = ok2 ? xrb[idx2] : 0.0f;
        sxi[256 + i] = ok2 ? xib[idx2] : 0.0f;
    }
    if (i < Tz) { sar[i] = ar[i]; sai[i] = ai[i]; }
    __syncthreads();

    float accr = 0.0f, acci = 0.0f;
    for (int t = 0; t < Tz; ++t) {
        const float wr = sar[t];
        const float wi = sai[t];
        const float vr = sxr[i + 128 - t];   // x[b, n + h - t]
        const float vi = sxi[i + 128 - t];
        accr = fmaf(wr, vr, accr);
        accr = fmaf(-wi, vi, accr);
        acci = fmaf(wr, vi, acci);
        acci = fmaf(wi, vr, acci);
    }

    const size_t o = ((size_t)b * Sz + n0 + i) * 2;
    out[o]     = accr;
    out[o + 1] = acci;
}

// ---------------------------------------------------------------------------
// Kernel 2: colored-noise GEMM, fused real+imag against one stream of L.
//   nr[b,s] = scale * sum_t L[s,t] * noise_r[b,t]   (same for ni)
// One wave (32 threads) per 16-column tile: full M=64 strip, full K=8192.
// f32 WMMA 16x16x4; L streamed non-temporal (one-shot 256MB, keep L2 for the
// 4MB noise matrix); epilogue does out[...] += scale*acc (after FIR kernel).
// ---------------------------------------------------------------------------
__global__ __launch_bounds__(32) void noise_gemm_kernel(
    const float* __restrict__ Lm, const float* __restrict__ nr,
    const float* __restrict__ ni, const float* __restrict__ n0p,
    float* __restrict__ out)
{
    const int lane = threadIdx.x & 31;
    const int half = lane >> 4;      // 0: K=0,1  1: K=2,3
    const int l    = lane & 15;      // row (A) / column (B) within tile
    const int s0   = blockIdx.x * 16;
    const int col  = s0 + l;

    // B fragment source: Bm[k][n] = L[s0+n][k]  (L row-major [S][S])
    const float* Lrow   = Lm + (size_t)col * Sz + 2 * half;
    // A fragment sources: noise[mt*16 + l][k]
    const float* ArBase = nr + (size_t)l * Sz + 2 * half;
    const float* AiBase = ni + (size_t)l * Sz + 2 * half;

    v8f accR[4] = {};
    v8f accI[4] = {};

    for (int kc = 0; kc < Sz; kc += 128) {
        // keep the 256MB L stream ahead of the WMMA pipe; once per chunk
        if (kc + 640 <= Sz) {
            __builtin_prefetch(Lrow + kc + 512, 0, 0);
        }
#pragma unroll 4
        for (int k0 = kc; k0 < kc + 128; k0 += 4) {
            // one-shot stream: non-temporal so L never evicts hot A / out data
            const v2f bf = __builtin_nontemporal_load(
                reinterpret_cast<const v2f*>(Lrow + k0));
#pragma unroll
            for (int mt = 0; mt < 4; ++mt) {
                const v2f arf = *reinterpret_cast<const v2f*>(ArBase + (size_t)mt * 16 * Sz + k0);
                const v2f aif = *reinterpret_cast<const v2f*>(AiBase + (size_t)mt * 16 * Sz + k0);
                accR[mt] = __builtin_amdgcn_wmma_f32_16x16x4_f32(
                    false, arf, false, bf, (short)0, accR[mt], false, false);
                accI[mt] = __builtin_amdgcn_wmma_f32_16x16x4_f32(
                    false, aif, false, bf, (short)0, accI[mt], false, false);
            }
        }
    }

    // scale is linear -> applied at epilogue; computed from N0 on device
    const float n0db  = n0p[0];
    const float scale = sqrtf(0.5f * __expf(n0db * 0.23025850929940457f)); // 10^(x/10)

    // C/D layout: VGPR j -> M = j + 8*half (per lane half), N = col
#pragma unroll
    for (int mt = 0; mt < 4; ++mt) {
#pragma unroll
        for (int j = 0; j < 8; ++j) {
            const int row = mt * 16 + half * 8 + j;
            float* po = out + ((size_t)row * Sz + col) * 2;
            v2f cur = *reinterpret_cast<const v2f*>(po);
            cur.x += scale * accR[mt][j];
            cur.y += scale * accI[mt][j];
            *reinterpret_cast<v2f*>(po) = cur;
        }
    }
}

// ---------------------------------------------------------------------------
extern "C" void kernel_launch(void* const* d_in, const int* in_sizes, int n_in,
                              void* d_out, int out_size, void* d_ws, size_t ws_size,
                              hipStream_t stream)
{
    (void)in_sizes; (void)n_in; (void)out_size; (void)d_ws; (void)ws_size;

    const float* xr = (const float*)d_in[0];
    const float* xi = (const float*)d_in[1];
    const float* ar = (const float*)d_in[2];
    const float* ai = (const float*)d_in[3];
    const float* Lm = (const float*)d_in[4];
    const float* nr = (const float*)d_in[5];
    const float* ni = (const float*)d_in[6];
    const float* n0 = (const float*)d_in[7];
    float* out = (float*)d_out;

    // 1) FIR writes out[B,S,2]
    fir_kernel<<<dim3(Sz / 256, Bz), 256, 0, stream>>>(xr, xi, ar, ai, out);
    // 2) GEMM adds colored noise into out (stream-ordered after FIR)
    noise_gemm_kernel<<<dim3(Sz / 16), 32, 0, stream>>>(Lm, nr, ni, n0, out);
}